// SurfaceDiceLoss_69973607186631
// MI455X (gfx1250) — compile-verified
//
#include <hip/hip_runtime.h>
#include <math.h>

// Problem dims (fixed by setup_inputs): pred (1,32,256,256) f32, labels same i32, area (256,) f32.
#define ZDIM 32
#define YDIM 256
#define XDIM 256
#define ZPAIRS (ZDIM - 1)          // 31
#define YO (YDIM + 1)              // 257
#define XO (XDIM + 1)              // 257
#define LPP (YO * XO)              // 66049 cubes per pair
#define NCUBES (ZPAIRS * LPP)      // 2047519
#define SMOOTHF 1e-3f
#define BLK 256

typedef __attribute__((ext_vector_type(2))) float v2f;
typedef __attribute__((ext_vector_type(8))) float v8f;

// Full-wave (32 lane) sum via V_WMMA_F32_16X16X4_F32.
// A (16x4 f32, 2 VGPRs): lanes 0-15 -> rows M=0..15 with {K0,K1}, lanes 16-31 -> {K2,K3}.
// We put v in .x (K0 / K2) and 0 in .y, B = all ones -> D[m][n] = v_m + v_{m+16} for all n.
// Lane n (n<16) holds D[0..7][n] in its 8 dest VGPRs; lane 16+n holds D[8..15][n].
// Row-sum the 8 VGPRs, then one xor-16 shuffle gives the complete 32-lane sum in EVERY lane.
// Requires EXEC == all ones at the call site.
__device__ __forceinline__ float wave_sum_wmma(float v) {
    v2f a; a.x = v;   a.y = 0.0f;
    v2f b; b.x = 1.0f; b.y = 1.0f;
    v8f c = {};
    v8f d = __builtin_amdgcn_wmma_f32_16x16x4_f32(
        /*neg_a=*/false, a, /*neg_b=*/false, b,
        /*c_mod=*/(short)0, c, /*reuse_a=*/false, /*reuse_b=*/false);
    float s = ((d[0] + d[1]) + (d[2] + d[3])) + ((d[4] + d[5]) + (d[6] + d[7]));
    s += __shfl_xor(s, 16, 32);
    return s;
}

__global__ void __launch_bounds__(BLK)
surface_dice_cubes(const float* __restrict__ pred,
                   const int*   __restrict__ labels,
                   const float* __restrict__ area,
                   float*       __restrict__ partial)   // [2 * gridDim.x]
{
    __shared__ float s_area[256];
    __shared__ float s_red[2 * (BLK / 32)];

    const int t = threadIdx.x;
    s_area[t] = area[t];                 // BLK == 256 == table size
    __syncthreads();

    // ---- map thread -> cube; keep ALL lanes live (WMMA needs full EXEC) ----
    long long gidx = (long long)blockIdx.x * BLK + t;
    const bool valid = gidx < (long long)NCUBES;
    const int  ii = valid ? (int)gidx : 0;
    const int  x  = ii % XO;
    const int  yq = ii / XO;
    const int  y  = yq % YO;
    const int  zp = yq / YO;

    // ---- gather the 8 cube corners (zero outside; pad is applied AFTER sigmoid) ----
    float r[8];
    int lcode = 0;
#pragma unroll
    for (int c = 0; c < 2; ++c) {
        const int z = zp + c;
#pragma unroll
        for (int i = 0; i < 2; ++i) {
            const int gy = y + i - 1;
#pragma unroll
            for (int j = 0; j < 2; ++j) {
                const int gx = x + j - 1;
                const int k  = c * 4 + i * 2 + j;       // bit index, matches reference
                float p = 0.0f;
                int   lb = 0;
                if (gy >= 0 && gy < YDIM && gx >= 0 && gx < XDIM) {
                    const int off = (z * YDIM + gy) * XDIM + gx;
                    const float v = pred[off];
                    p  = 1.0f / (1.0f + expf(-v));      // sigmoid, strictly > 0
                    lb = labels[off] & 1;
                }
                r[k] = p;
                lcode |= lb << k;
            }
        }
    }

    // ---- 8-step greedy cube decomposition (branchless, fully unrolled) ----
    float pred_area   = 0.0f;
    float pw_in_label = 0.0f;
#pragma unroll
    for (int it = 0; it < 8; ++it) {
        int   code = 0;
        float w    = __builtin_inff();
#pragma unroll
        for (int k = 0; k < 8; ++k) {
            const bool nz = r[k] > 0.0f;
            code |= ((int)nz) << k;
            w = fminf(w, nz ? r[k] : __builtin_inff());
        }
        w = (code != 0) ? w : 0.0f;                     // avoid inf (and 0*inf=NaN)
        pred_area   += s_area[code] * w;                // LDS gather (ds_load)
        pw_in_label += (code == lcode) ? w : 0.0f;
#pragma unroll
        for (int k = 0; k < 8; ++k)
            r[k] -= (r[k] > 0.0f) ? w : 0.0f;           // min corner lands exactly on 0
    }

    const float la   = s_area[lcode];
    const float cnum = valid ? (2.0f * pw_in_label * la) : 0.0f;
    const float cden = valid ? (la + pred_area)          : 0.0f;

    // ---- per-wave WMMA reduction, then per-block combine ----
    const float wnum = wave_sum_wmma(cnum);
    const float wden = wave_sum_wmma(cden);
    if ((t & 31) == 0) {
        const int wid = t >> 5;
        s_red[2 * wid]     = wnum;
        s_red[2 * wid + 1] = wden;
    }
    __syncthreads();
    if (t == 0) {
        float bn = 0.0f, bd = 0.0f;
#pragma unroll
        for (int wv = 0; wv < BLK / 32; ++wv) {
            bn += s_red[2 * wv];
            bd += s_red[2 * wv + 1];
        }
        partial[2 * blockIdx.x]     = bn;
        partial[2 * blockIdx.x + 1] = bd;
    }
}

__global__ void __launch_bounds__(32)
surface_dice_final(const float* __restrict__ partial, int nblocks,
                   float* __restrict__ out)
{
    const int lane = threadIdx.x;                       // exactly one wave32
    float n = 0.0f, d = 0.0f;
    for (int i = lane; i < nblocks; i += 32) {
        n += partial[2 * i];
        d += partial[2 * i + 1];
    }
    const float tn = wave_sum_wmma(n);                  // full EXEC: all 32 lanes live
    const float td = wave_sum_wmma(d);
    if (lane == 0)
        out[0] = 1.0f - (tn + SMOOTHF) / (td + SMOOTHF);
}

extern "C" void kernel_launch(void* const* d_in, const int* in_sizes, int n_in,
                              void* d_out, int out_size, void* d_ws, size_t ws_size,
                              hipStream_t stream) {
    (void)in_sizes; (void)n_in; (void)out_size; (void)ws_size;
    const float* pred   = (const float*)d_in[0];
    const int*   labels = (const int*)  d_in[1];
    const float* area   = (const float*)d_in[2];
    float* out     = (float*)d_out;
    float* partial = (float*)d_ws;                      // 2 floats per block, all slots written

    const int blocks = (NCUBES + BLK - 1) / BLK;        // 7999 -> 64 KB of d_ws
    surface_dice_cubes<<<blocks, BLK, 0, stream>>>(pred, labels, area, partial);
    surface_dice_final<<<1, 32, 0, stream>>>(partial, blocks, out);
}